// EMALayer_50955492000597
// MI455X (gfx1250) — compile-verified
//
#include <hip/hip_runtime.h>
#include <hip/hip_bf16.h>

// EMA layer (MEGA-style) for MI455X / gfx1250.
// Chunked state-space evaluation of the length-4096 causal conv:
//   k_d[l] = sum_n c_n q_n^l ;  out = conv(k, x) + omega * x
// Intra-chunk Toeplitz matmul done with V_WMMA_F32_16X16X4_F32 (f32, exact).

typedef float v2f __attribute__((ext_vector_type(2)));
typedef float v8f __attribute__((ext_vector_type(8)));

#define SEQ_LEN 4096
#define BSZ     8
#define EMBED   1024
#define NDIM    16
#define CHUNK   64
#define NCHUNK  (SEQ_LEN / CHUNK)   // 64
#define NPAIR   (NCHUNK / 2)        // 32
#define DTILE   8                   // channels per workgroup in main kernel

// ---------------------------------------------------------------------------
// K1: per-channel parameters: q, c = p*beta*gamma, ln(q), q^64, and the
// Toeplitz column kcol[d][m] = sum_n c_n q_n^m  (m = 0..63).
// Same exp(t * log q) construction as the reference.
// grid = EMBED blocks, 64 threads (thread t -> kcol entry m = t).
// ---------------------------------------------------------------------------
__global__ void ema_prep(const float* __restrict__ delta,
                         const float* __restrict__ alpha,
                         const float* __restrict__ beta,
                         const float* __restrict__ gamma,
                         float* __restrict__ qArr,  float* __restrict__ cArr,
                         float* __restrict__ lqArr, float* __restrict__ qCArr,
                         float* __restrict__ kcol) {
  const int d = blockIdx.x;
  const int t = threadIdx.x;                // 0..63
  const float dd = __expf(delta[d]);
  float qn[NDIM], cn[NDIM];
#pragma unroll
  for (int n = 0; n < NDIM; ++n) {
    const float al = alpha[d * NDIM + n];
    const float p  = dd / (1.0f + 0.5f * dd * al);
    qn[n] = 1.0f - p * al;                  // in (0,1) by construction
    cn[n] = p * beta[d * NDIM + n] * gamma[d * NDIM + n];
  }
  float s = 0.0f;
#pragma unroll
  for (int n = 0; n < NDIM; ++n)
    s += cn[n] * __expf((float)t * __logf(qn[n]));
  kcol[d * CHUNK + t] = s;

  if (t < NDIM) {
    const float q = qn[t];
    qArr [d * NDIM + t] = q;
    cArr [d * NDIM + t] = cn[t];
    lqArr[d * NDIM + t] = __logf(q);
    float qc = q;                           // q^64 by repeated squaring
#pragma unroll
    for (int i = 0; i < 6; ++i) qc *= qc;
    qCArr[d * NDIM + t] = qc;
  }
}

// ---------------------------------------------------------------------------
// K2: per (chunk, b, d): chunk-local partial state A_n = sum_j q^{63-j} x_j
// (Horner). x reads coalesced over d (innermost dim of (L,B,D)).
// Abuf layout: [chunk][b][n][d].
// grid = (NCHUNK, BSZ, EMBED/256), block = 256.
// ---------------------------------------------------------------------------
__global__ void ema_chunkstate(const float* __restrict__ x,
                               const float* __restrict__ qArr,
                               float* __restrict__ Abuf) {
  const int ch = blockIdx.x;
  const int b  = blockIdx.y;
  const int d  = blockIdx.z * blockDim.x + threadIdx.x;

  float q[NDIM], A[NDIM];
#pragma unroll
  for (int n = 0; n < NDIM; ++n) { q[n] = qArr[d * NDIM + n]; A[n] = 0.0f; }

  const float* xp = x + ((size_t)ch * CHUNK * BSZ + b) * EMBED + d;
  for (int j = 0; j < CHUNK; ++j) {
    const float xv = xp[(size_t)j * BSZ * EMBED];
#pragma unroll
    for (int n = 0; n < NDIM; ++n) A[n] = A[n] * q[n] + xv;
  }
  const size_t base = (((size_t)ch * BSZ + b) * NDIM) * EMBED + d;
#pragma unroll
  for (int n = 0; n < NDIM; ++n) Abuf[base + (size_t)n * EMBED] = A[n];
}

// ---------------------------------------------------------------------------
// K3: sequential scan over chunks (in-place): Abuf[ch] becomes the state
// *entering* chunk ch:  S(ch) = q^64 * S(ch-1) + A(ch-1),  S(0) = 0.
// One thread per (b,n,d); fully coalesced.
// grid = BSZ*NDIM*EMBED/256, block = 256.
// ---------------------------------------------------------------------------
__global__ void ema_scan(const float* __restrict__ qCArr,
                         float* __restrict__ Abuf) {
  const int flat = blockIdx.x * blockDim.x + threadIdx.x;  // (b*N + n)*D + d
  const int d = flat % EMBED;
  const int n = (flat / EMBED) % NDIM;
  const float qc = qCArr[d * NDIM + n];
  float S = 0.0f;
  const size_t stride = (size_t)BSZ * NDIM * EMBED;
  for (int ch = 0; ch < NCHUNK; ++ch) {
    const size_t idx = (size_t)ch * stride + flat;
    const float a = Abuf[idx];
    Abuf[idx] = S;
    S = qc * S + a;
  }
}

// ---------------------------------------------------------------------------
// K4: main kernel. Workgroup = (DTILE=8 channels) x (one chunk pair).
// WMMA N dimension packs 16 columns = 2 chunks x 8 batches (full tiles).
// Per channel d:  out(64x16) = T(64x64) @ X(64x16)  +  W(64x16) @ S(16x16)
// with T[i][j] = kcol[i-j] (causal), W[i][n] = c_n q_n^{i+1}.
// f32 16x16x4 WMMA; 4 waves (128 threads), wave w owns rows [16w,16w+16).
// X buffer reused in-place as output stage; residual folded in.
// grid = (EMBED/DTILE, NPAIR), block = 128.
// ---------------------------------------------------------------------------
__global__ __launch_bounds__(128) void ema_main(
    const float* __restrict__ x,     const float* __restrict__ omega,
    const float* __restrict__ cArr,  const float* __restrict__ lqArr,
    const float* __restrict__ kcol,  const float* __restrict__ Sbuf,
    float* __restrict__ out) {
  // 32KB + 8KB + 16.5KB + 4.25KB = ~61KB LDS
  __shared__ __align__(16) float Xall[128 * BSZ * DTILE];       // [t][b][dl]
  __shared__ __align__(16) float Sl[NDIM * DTILE * 16];          // [n][dl][col]
  __shared__ __align__(16) float Tm[64 * 66];                    // padded rows
  __shared__ __align__(16) float Wm[64 * 17];

  const int tid  = threadIdx.x;
  const int d0   = blockIdx.x * DTILE;
  const int pair = blockIdx.y;
  const int t0   = pair * 128;

  // Stage X tile (128 t x 8 b x 8 dl) with coalesced float4 loads.
  for (int e = tid; e < 2048; e += 128) {
    const int t = e >> 4, rem = e & 15, b = rem >> 1, v = rem & 1;
    const float4 val =
        *(const float4*)(x + (((size_t)(t0 + t) * BSZ + b) * EMBED + d0 + v * 4));
    *(float4*)&Xall[(t * BSZ + b) * DTILE + v * 4] = val;
  }
  // Stage incoming states: Sl[n][dl][col], col = cp*8 + b.
  for (int e = tid; e < NDIM * DTILE * 16; e += 128) {
    const int col = e >> 7, n = (e >> 3) & 15, dl = e & 7;
    const int cp = col >> 3, b = col & 7;
    const int ch = 2 * pair + cp;
    Sl[(n * DTILE + dl) * 16 + col] =
        Sbuf[(((size_t)ch * BSZ + b) * NDIM + n) * EMBED + d0 + dl];
  }

  const int lane = tid & 31;
  const int wv   = tid >> 5;
  const int c    = lane & 15;          // WMMA column (A row index / B-D col)
  const int kk   = (lane >> 4) << 1;   // K sub-offset for upper half-wave
  const int hi8  = (lane >> 4) << 3;   // C/D row offset for upper half-wave
  const int i0   = wv * 16;
  const int cp   = c >> 3, cb = c & 7; // column -> (chunk-in-pair, batch)

  for (int dl = 0; dl < DTILE; ++dl) {
    __syncthreads();
    const int d = d0 + dl;
    // Build causal Toeplitz matrix T (row stride 66 to dodge bank conflicts).
    for (int e = tid; e < 64 * 64; e += 128) {
      const int i = e >> 6, j = e & 63;
      Tm[i * 66 + j] = (j <= i) ? kcol[d * CHUNK + (i - j)] : 0.0f;
    }
    // Build state-injection matrix W[i][n] = c_n * q_n^(i+1).
    for (int e = tid; e < 64 * NDIM; e += 128) {
      const int i = e >> 4, n = e & 15;
      Wm[i * 17 + n] = cArr[d * NDIM + n] *
                       __expf((float)(i + 1) * lqArr[d * NDIM + n]);
    }
    __syncthreads();

    v8f acc = {0.f, 0.f, 0.f, 0.f, 0.f, 0.f, 0.f, 0.f};
    // W(64x16) @ S(16x16): K = 16 -> 4 WMMA steps.
#pragma unroll
    for (int k = 0; k < 16; k += 4) {
      v2f a, b;
      a.x = Wm[(i0 + c) * 17 + k + kk];
      a.y = Wm[(i0 + c) * 17 + k + kk + 1];
      b.x = Sl[((k + kk    ) * DTILE + dl) * 16 + c];
      b.y = Sl[((k + kk + 1) * DTILE + dl) * 16 + c];
      acc = __builtin_amdgcn_wmma_f32_16x16x4_f32(false, a, false, b,
                                                  (short)0, acc, false, false);
    }
    // T(64x64) @ X(64x16): K = 64 -> 16 WMMA steps.
#pragma unroll
    for (int k = 0; k < 64; k += 4) {
      v2f a, b;
      a.x = Tm[(i0 + c) * 66 + k + kk];
      a.y = Tm[(i0 + c) * 66 + k + kk + 1];
      const int t1 = cp * 64 + k + kk;
      b.x = Xall[((t1    ) * BSZ + cb) * DTILE + dl];
      b.y = Xall[((t1 + 1) * BSZ + cb) * DTILE + dl];
      acc = __builtin_amdgcn_wmma_f32_16x16x4_f32(false, a, false, b,
                                                  (short)0, acc, false, false);
    }

    __syncthreads();  // everyone done reading Xall[.][.][dl] before overwrite
    const float om = omega[d];
#pragma unroll
    for (int r = 0; r < 8; ++r) {
      const int i  = i0 + r + hi8;     // row within chunk (C/D layout)
      const int t1 = cp * 64 + i;
      const int idx = (t1 * BSZ + cb) * DTILE + dl;
      const float xv = Xall[idx];      // same lane read->write: safe in-place
      Xall[idx] = acc[r] + om * xv;    // residual folded in
    }
  }
  __syncthreads();
  // Coalesced float4 store of the (now output) tile.
  for (int e = tid; e < 2048; e += 128) {
    const int t = e >> 4, rem = e & 15, b = rem >> 1, v = rem & 1;
    const float4 val = *(const float4*)&Xall[(t * BSZ + b) * DTILE + v * 4];
    *(float4*)(out + (((size_t)(t0 + t) * BSZ + b) * EMBED + d0 + v * 4)) = val;
  }
}

// ---------------------------------------------------------------------------
extern "C" void kernel_launch(void* const* d_in, const int* in_sizes, int n_in,
                              void* d_out, int out_size, void* d_ws,
                              size_t ws_size, hipStream_t stream) {
  const float* x     = (const float*)d_in[0];
  const float* delta = (const float*)d_in[1];
  const float* alpha = (const float*)d_in[2];
  const float* beta  = (const float*)d_in[3];
  const float* gamma = (const float*)d_in[4];
  const float* omega = (const float*)d_in[5];
  float* out = (float*)d_out;

  // Workspace layout (floats): params + chunk-state buffer (~34.1 MB total).
  float* ws    = (float*)d_ws;
  float* qArr  = ws;                         // EMBED*NDIM
  float* cArr  = qArr  + EMBED * NDIM;       // EMBED*NDIM
  float* lqArr = cArr  + EMBED * NDIM;       // EMBED*NDIM
  float* qCArr = lqArr + EMBED * NDIM;       // EMBED*NDIM
  float* kcol  = qCArr + EMBED * NDIM;       // EMBED*CHUNK
  float* Abuf  = kcol  + EMBED * CHUNK;      // NCHUNK*BSZ*NDIM*EMBED (32 MB)

  ema_prep<<<EMBED, CHUNK, 0, stream>>>(delta, alpha, beta, gamma,
                                        qArr, cArr, lqArr, qCArr, kcol);
  ema_chunkstate<<<dim3(NCHUNK, BSZ, EMBED / 256), 256, 0, stream>>>(x, qArr,
                                                                     Abuf);
  ema_scan<<<(BSZ * NDIM * EMBED) / 256, 256, 0, stream>>>(qCArr, Abuf);
  ema_main<<<dim3(EMBED / DTILE, NPAIR), 128, 0, stream>>>(
      x, omega, cArr, lqArr, kcol, Abuf, out);
}